// Output_2130303779367
// MI455X (gfx1250) — compile-verified
//
#include <hip/hip_runtime.h>

// Frame interpolation blend (SuperSloMo-style):
//   g0 = dense_image_warp(frames_0, f_t0); g1 = dense_image_warp(frames_1, f_t1)
//   out = (w0*g0 + w1*g1) / (w0 + w1 + 1e-12), w0=(1-t)*v0, w1=t*v1
// HBM-gather-bound; no matmul -> WMMA inapplicable. gfx1250 levers:
//   - th:TH_NT non-temporal hints on streamed tensors (preserve 192MB L2 for
//     the gathered frames, which have ~4x tap reuse)
//   - compile-time 512x512 specialization: shifts/masks instead of int div,
//     block-uniform batch index -> scalar s_load of t.

#define DEVINL __device__ __forceinline__

struct F3 { float c0, c1, c2; };

DEVINL float nt_load(const float* p) { return __builtin_nontemporal_load(p); }

// Bilinear sample of a [H,W,3] fp32 image at (qy, qx), tfa-style clamping.
template <int H, int W>
DEVINL F3 bilinear3(const float* __restrict__ img, float qy, float qx) {
    float fy = fminf(fmaxf(floorf(qy), 0.0f), (float)(H - 2));
    float fx = fminf(fmaxf(floorf(qx), 0.0f), (float)(W - 2));
    float ay = fminf(fmaxf(qy - fy, 0.0f), 1.0f);
    float ax = fminf(fmaxf(qx - fx, 0.0f), 1.0f);
    int iy = (int)fy;
    int ix = (int)fx;

    const float* p0 = img + (iy * W + ix) * 3;   // top row: tl(3) tr(3)
    const float* p1 = p0 + W * 3;                // bottom row: bl(3) br(3)

    // 6 consecutive floats per row -> backend merges into b128+b64 loads.
    float tl0 = p0[0], tl1 = p0[1], tl2 = p0[2];
    float tr0 = p0[3], tr1 = p0[4], tr2 = p0[5];
    float bl0 = p1[0], bl1 = p1[1], bl2 = p1[2];
    float br0 = p1[3], br1 = p1[4], br2 = p1[5];

    float top0 = fmaf(ax, tr0 - tl0, tl0);
    float top1 = fmaf(ax, tr1 - tl1, tl1);
    float top2 = fmaf(ax, tr2 - tl2, tl2);
    float bot0 = fmaf(ax, br0 - bl0, bl0);
    float bot1 = fmaf(ax, br1 - bl1, bl1);
    float bot2 = fmaf(ax, br2 - bl2, bl2);

    F3 r;
    r.c0 = fmaf(ay, bot0 - top0, top0);
    r.c1 = fmaf(ay, bot1 - top1, top1);
    r.c2 = fmaf(ay, bot2 - top2, top2);
    return r;
}

// Specialized kernel: H=W=512 compile-time, blockDim=256.
// Pixels per image = 2^18; blocks per image = 2^10 -> b is block-uniform.
template <int H, int W>
__global__ __launch_bounds__(256) void warp_blend_pow2_kernel(
    const float* __restrict__ frames0, const float* __restrict__ ft0,
    const float* __restrict__ vt0,     const float* __restrict__ frames1,
    const float* __restrict__ ft1,     const float* __restrict__ vt1,
    const float* __restrict__ tind,    float* __restrict__ out, int N) {
    constexpr int HW = H * W;                    // power of two
    int idx = blockIdx.x * 256 + threadIdx.x;
    if (idx >= N) return;

    int x = idx & (W - 1);
    int y = (idx / W) & (H - 1);
    int b = (int)(blockIdx.x / (HW / 256));      // block-uniform -> SGPR

    // Streamed-once data: non-temporal (th:NT) loads.
    float f0y = nt_load(ft0 + 2 * idx);
    float f0x = nt_load(ft0 + 2 * idx + 1);
    float f1y = nt_load(ft1 + 2 * idx);
    float f1x = nt_load(ft1 + 2 * idx + 1);
    float v0  = nt_load(vt0 + idx);
    float v1  = nt_load(vt1 + idx);
    float t   = tind[b];                         // uniform -> s_load

    const float* img0 = frames0 + (long)b * HW * 3;
    const float* img1 = frames1 + (long)b * HW * 3;

    F3 g0 = bilinear3<H, W>(img0, (float)y - f0y, (float)x - f0x);
    F3 g1 = bilinear3<H, W>(img1, (float)y - f1y, (float)x - f1x);

    float w0 = (1.0f - t) * v0;
    float w1 = t * v1;
    float rz = 1.0f / (w0 + w1 + 1e-12f);

    float o0 = (w0 * g0.c0 + w1 * g1.c0) * rz;
    float o1 = (w0 * g0.c1 + w1 * g1.c1) * rz;
    float o2 = (w0 * g0.c2 + w1 * g1.c2) * rz;

    float* po = out + 3 * (long)idx;
    __builtin_nontemporal_store(o0, po + 0);
    __builtin_nontemporal_store(o1, po + 1);
    __builtin_nontemporal_store(o2, po + 2);
}

// Generic fallback (runtime H, W) — same math, used only if shape differs.
__global__ __launch_bounds__(256) void warp_blend_generic_kernel(
    const float* __restrict__ frames0, const float* __restrict__ ft0,
    const float* __restrict__ vt0,     const float* __restrict__ frames1,
    const float* __restrict__ ft1,     const float* __restrict__ vt1,
    const float* __restrict__ tind,    float* __restrict__ out,
    int H, int W, int N) {
    int idx = blockIdx.x * blockDim.x + threadIdx.x;
    if (idx >= N) return;

    int x  = idx % W;
    int t2 = idx / W;
    int y  = t2 % H;
    int b  = t2 / H;

    float f0y = nt_load(ft0 + 2 * (long)idx);
    float f0x = nt_load(ft0 + 2 * (long)idx + 1);
    float f1y = nt_load(ft1 + 2 * (long)idx);
    float f1x = nt_load(ft1 + 2 * (long)idx + 1);
    float v0  = nt_load(vt0 + idx);
    float v1  = nt_load(vt1 + idx);
    float t   = tind[b];

    const float* img0 = frames0 + (long)b * H * W * 3;
    const float* img1 = frames1 + (long)b * H * W * 3;

    // Inline generic bilinear (runtime strides).
    auto bil = [&](const float* img, float qy, float qx, F3& r) {
        float fy = fminf(fmaxf(floorf(qy), 0.0f), (float)(H - 2));
        float fx = fminf(fmaxf(floorf(qx), 0.0f), (float)(W - 2));
        float ay = fminf(fmaxf(qy - fy, 0.0f), 1.0f);
        float ax = fminf(fmaxf(qx - fx, 0.0f), 1.0f);
        int iy = (int)fy, ix = (int)fx;
        const float* p0 = img + ((long)iy * W + ix) * 3;
        const float* p1 = p0 + (long)W * 3;
        float tl0 = p0[0], tl1 = p0[1], tl2 = p0[2];
        float tr0 = p0[3], tr1 = p0[4], tr2 = p0[5];
        float bl0 = p1[0], bl1 = p1[1], bl2 = p1[2];
        float br0 = p1[3], br1 = p1[4], br2 = p1[5];
        float top0 = fmaf(ax, tr0 - tl0, tl0);
        float top1 = fmaf(ax, tr1 - tl1, tl1);
        float top2 = fmaf(ax, tr2 - tl2, tl2);
        float bot0 = fmaf(ax, br0 - bl0, bl0);
        float bot1 = fmaf(ax, br1 - bl1, bl1);
        float bot2 = fmaf(ax, br2 - bl2, bl2);
        r.c0 = fmaf(ay, bot0 - top0, top0);
        r.c1 = fmaf(ay, bot1 - top1, top1);
        r.c2 = fmaf(ay, bot2 - top2, top2);
    };

    F3 g0, g1;
    bil(img0, (float)y - f0y, (float)x - f0x, g0);
    bil(img1, (float)y - f1y, (float)x - f1x, g1);

    float w0 = (1.0f - t) * v0;
    float w1 = t * v1;
    float rz = 1.0f / (w0 + w1 + 1e-12f);

    float o0 = (w0 * g0.c0 + w1 * g1.c0) * rz;
    float o1 = (w0 * g0.c1 + w1 * g1.c1) * rz;
    float o2 = (w0 * g0.c2 + w1 * g1.c2) * rz;

    float* po = out + 3 * (long)idx;
    __builtin_nontemporal_store(o0, po + 0);
    __builtin_nontemporal_store(o1, po + 1);
    __builtin_nontemporal_store(o2, po + 2);
}

extern "C" void kernel_launch(void* const* d_in, const int* in_sizes, int n_in,
                              void* d_out, int out_size, void* d_ws, size_t ws_size,
                              hipStream_t stream) {
    const float* frames0 = (const float*)d_in[0];
    const float* ft0     = (const float*)d_in[1];
    const float* vt0     = (const float*)d_in[2];
    const float* frames1 = (const float*)d_in[3];
    const float* ft1     = (const float*)d_in[4];
    const float* vt1     = (const float*)d_in[5];
    const float* tind    = (const float*)d_in[6];
    float* out           = (float*)d_out;

    int B  = in_sizes[6];            // t_indeces: [B,1,1,1]
    int N  = in_sizes[2];            // v_t0: B*H*W
    int HW = N / (B > 0 ? B : 1);

    int threads = 256;
    int blocks  = (N + threads - 1) / threads;

    if (HW == 512 * 512) {
        warp_blend_pow2_kernel<512, 512><<<blocks, threads, 0, stream>>>(
            frames0, ft0, vt0, frames1, ft1, vt1, tind, out, N);
    } else {
        int W = 512;
        if (HW % 512 != 0) W = HW;
        int H = HW / W;
        warp_blend_generic_kernel<<<blocks, threads, 0, stream>>>(
            frames0, ft0, vt0, frames1, ft1, vt1, tind, out, H, W, N);
    }
}